// LengthRegulator_8323646620288
// MI455X (gfx1250) — compile-verified
//
#include <hip/hip_runtime.h>
#include <cstdint>

// LengthRegulator for MI455X (gfx1250, wave32).
// Pure gather/scatter: HBM-write-bound (~201 MB out @ 23.3 TB/s => ~9us floor).
// CDNA5 paths used: global_load_async_to_lds_b128 + s_wait_asynccnt for LDS
// staging of the duration row, ds_* for the in-LDS scan/search, b128 vector
// moves, nontemporal (TH=NT) streaming stores for the write-once output.

typedef float v4f __attribute__((ext_vector_type(4)));

constexpr int B_DIM        = 32;
constexpr int T_DIM        = 512;
constexpr int D_DIM        = 384;
constexpr int MAX_LEN      = 4096;          // T * DUR_MAX
constexpr int FRAMES       = 32;            // output frames per block
constexpr int BLOCK        = 256;           // 8 waves of 32
constexpr int V4_PER_FRAME = D_DIM / 4;     // 96 float4 per frame
constexpr int BLOCKS_PER_ROW = MAX_LEN / FRAMES; // 128

__global__ __launch_bounds__(BLOCK)
void length_regulator_kernel(const float* __restrict__ x,
                             const float* __restrict__ dur,
                             float* __restrict__ out) {
    __shared__ int s_cum[T_DIM];     // staged dur bits -> repeats -> inclusive cumsum
    __shared__ int s_idx[FRAMES];    // source token per output frame (-1 = pad)

    const int tid = threadIdx.x;
    const int b   = blockIdx.x / BLOCKS_PER_ROW;
    const int j0  = (blockIdx.x % BLOCKS_PER_ROW) * FRAMES;

    // ---- Stage dur[b, 0:512] (2 KB) into LDS via gfx1250 async copy engine.
    // Threads 0..127 each move 16 bytes: 4 waves x (32 lanes x b128).
    if (tid < T_DIM / 4) {
        uint32_t lds_off = (uint32_t)(uintptr_t)(&s_cum[tid * 4]);
        uint64_t gaddr   = (uint64_t)(uintptr_t)(dur + (size_t)b * T_DIM + tid * 4);
        asm volatile("global_load_async_to_lds_b128 %0, %1, off"
                     :
                     : "v"(lds_off), "v"(gaddr)
                     : "memory");
    }
    asm volatile("s_wait_asynccnt 0x0" ::: "memory");
    __syncthreads();

    // ---- dur (float bits) -> integer repeats, in place (2 elems/thread).
    for (int i = tid; i < T_DIM; i += BLOCK) {
        float f  = __int_as_float(s_cum[i]);
        s_cum[i] = (int)(fmaxf(f, 0.0f) + 0.5f);   // trunc == floor for f>=0
    }
    __syncthreads();

    // ---- Inclusive scan over 512 elems, Hillis-Steele, 2 elems/thread.
    for (int off = 1; off < T_DIM; off <<= 1) {
        const int i0 = tid, i1 = tid + BLOCK;
        int v0 = (i0 >= off) ? s_cum[i0 - off] : 0;
        int v1 = (i1 >= off) ? s_cum[i1 - off] : 0;
        __syncthreads();
        s_cum[i0] += v0;
        s_cum[i1] += v1;
        __syncthreads();
    }

    const int total = s_cum[T_DIM - 1];

    // ---- Binary search (searchsorted right) for this block's 32 frames.
    if (tid < FRAMES) {
        const int j = j0 + tid;
        int lo = 0, hi = T_DIM;
        while (lo < hi) {
            int m = (lo + hi) >> 1;
            if (s_cum[m] > j) hi = m; else lo = m + 1;
        }
        s_idx[tid] = (j < total) ? lo : -1;
    }
    __syncthreads();

    // ---- Gather/copy: 32 frames x 96 float4 = 3072 b128 moves per block.
    // x loads stay cached (tokens repeat ~3.5x, whole x fits in L2);
    // out stores are write-once streaming -> nontemporal hint.
    const float* __restrict__ xrow = x + (size_t)b * T_DIM * D_DIM;
    float* __restrict__       orow = out + ((size_t)b * MAX_LEN + j0) * (size_t)D_DIM;

    #pragma unroll 4
    for (int k = tid; k < FRAMES * V4_PER_FRAME; k += BLOCK) {
        const int fi  = k / V4_PER_FRAME;
        const int e   = k - fi * V4_PER_FRAME;
        const int idx = s_idx[fi];
        v4f v = {0.0f, 0.0f, 0.0f, 0.0f};
        if (idx >= 0) {
            v = ((const v4f*)(xrow + (size_t)idx * D_DIM))[e];
        }
        __builtin_nontemporal_store(v, ((v4f*)(orow + (size_t)fi * D_DIM)) + e);
    }
}

extern "C" void kernel_launch(void* const* d_in, const int* in_sizes, int n_in,
                              void* d_out, int out_size, void* d_ws, size_t ws_size,
                              hipStream_t stream) {
    (void)in_sizes; (void)n_in; (void)out_size; (void)d_ws; (void)ws_size;
    const float* x   = (const float*)d_in[0];   // [B, T, D] float32
    const float* dur = (const float*)d_in[1];   // [B, T]    float32
    float* out       = (float*)d_out;           // [B, MAX_LEN, D] float32

    dim3 grid(B_DIM * BLOCKS_PER_ROW);          // 4096 blocks
    dim3 block(BLOCK);                          // 256 threads = 8 waves
    length_regulator_kernel<<<grid, block, 0, stream>>>(x, dur, out);
}